// Mamba3Block_60198261621479
// MI455X (gfx1250) — compile-verified
//
#include <hip/hip_runtime.h>
#include <hip/hip_bf16.h>
#include <math.h>

typedef __attribute__((ext_vector_type(2))) float v2f;
typedef __attribute__((ext_vector_type(8))) float v8f;

// ---------------------------------------------------------------------------
// fp32 WMMA GEMM:  C[M x Nout] = X[M x K] * W[Nout x K]^T
// Block = 256 threads (8 waves), 64x64 output tile, K staged in LDS 16 at a
// time via async global->LDS copies (ASYNCcnt), consumed by
// V_WMMA_F32_16X16X4_F32 (exact fp32, matches the fp32 reference).
// ---------------------------------------------------------------------------
__global__ __launch_bounds__(256)
void gemm_xwt_f32(const float* __restrict__ X, const float* __restrict__ W,
                  float* __restrict__ C, int M, int K, int Nout)
{
    __shared__ float Xs[64][17];
    __shared__ float Ws[64][17];

    const int tid  = threadIdx.x;
    const int wave = tid >> 5;
    const int lane = tid & 31;
    const int hi   = lane >> 4;     // 0 or 1 (half-wave)
    const int l16  = lane & 15;

    const int m0 = blockIdx.x * 64;
    const int n0 = blockIdx.y * 64;

    const int mw = (wave >> 1) * 16;   // wave's M offset in tile (0,16,32,48)
    const int nw = (wave & 1) * 32;    // wave's N offset in tile (0,32)

    v8f acc0 = {0.f,0.f,0.f,0.f,0.f,0.f,0.f,0.f};
    v8f acc1 = {0.f,0.f,0.f,0.f,0.f,0.f,0.f,0.f};

    // 256 threads cooperatively move a 64x16 panel, 16B (one b128) each
    const int lrow = tid >> 2;         // 0..63
    const int lcol = (tid & 3) * 4;    // 0,4,8,12

    // LDS byte addresses for this thread's 16B slot (low 32 bits of the
    // generic pointer are the workgroup-relative LDS offset)
    const unsigned xs_lds = (unsigned)(size_t)&Xs[lrow][lcol];
    const unsigned ws_lds = (unsigned)(size_t)&Ws[lrow][lcol];

    // W row clamp: OOB columns produce garbage that is never stored
    const int nrow = (n0 + lrow < Nout) ? (n0 + lrow) : (Nout - 1);

    for (int kc = 0; kc < K; kc += 16) {
        {
            const float* xsrc = X + (size_t)(m0 + lrow) * K + kc + lcol;
            const float* wsrc = W + (size_t)nrow * K + kc + lcol;
            // async global -> LDS, tracked by ASYNCcnt
            asm volatile("global_load_async_to_lds_b128 %0, %1, off"
                         :: "v"(xs_lds),
                            "v"((unsigned long long)(size_t)xsrc)
                         : "memory");
            asm volatile("global_load_async_to_lds_b128 %0, %1, off"
                         :: "v"(ws_lds),
                            "v"((unsigned long long)(size_t)wsrc)
                         : "memory");
        }
        if (kc + 16 < K) {
            // hint the next X panel toward the caches (global_prefetch_b8)
            __builtin_prefetch((const void*)(X + (size_t)(m0 + lrow) * K + kc + 16), 0, 3);
        }
        // drain this wave's async copies, then rendezvous with the other waves
        asm volatile("s_wait_asynccnt 0x0" ::: "memory");
        __syncthreads();

        #pragma unroll
        for (int k0 = 0; k0 < 16; k0 += 4) {
            // A 16x4 fp32 fragment: lanes 0-15 rows M, VGPR0/1 = K0/K1; lanes 16-31 = K2/K3
            v2f a, b0, b1;
            a.x  = Xs[mw + l16][k0 + 2*hi + 0];
            a.y  = Xs[mw + l16][k0 + 2*hi + 1];
            // B 4x16 fp32 fragment (W^T): VGPR0 rows {0,2}, VGPR1 rows {1,3}
            b0.x = Ws[nw + l16][k0 + 2*hi + 0];
            b0.y = Ws[nw + l16][k0 + 2*hi + 1];
            b1.x = Ws[nw + 16 + l16][k0 + 2*hi + 0];
            b1.y = Ws[nw + 16 + l16][k0 + 2*hi + 1];
            acc0 = __builtin_amdgcn_wmma_f32_16x16x4_f32(
                false, a, false, b0, (short)0, acc0, false, false);
            acc1 = __builtin_amdgcn_wmma_f32_16x16x4_f32(
                false, a, false, b1, (short)0, acc1, false, false);
        }
        __syncthreads();
    }

    // D layout: VGPR r -> row (mw + r + 8*hi), col (nw + l16) / (+16 for acc1)
    #pragma unroll
    for (int r = 0; r < 8; ++r) {
        const int m  = m0 + mw + r + 8 * hi;
        const int n  = n0 + nw + l16;
        const int n2 = n + 16;
        if (n  < Nout) C[(size_t)m * Nout + n ] = acc0[r];
        if (n2 < Nout) C[(size_t)m * Nout + n2] = acc1[r];
    }
}

// ---------------------------------------------------------------------------
// dt = softplus(s0 + bias), lam = sigmoid(s1), cum_angle = cumsum(dt * s2).
// One thread per (b,h) sequence: 16 sequences, trivially cheap.
// ---------------------------------------------------------------------------
__global__ void scalars_cumsum(const float* __restrict__ sel,
                               const float* __restrict__ dt_bias,
                               float* __restrict__ dt_a, float* __restrict__ lam_a,
                               float* __restrict__ cs_a, float* __restrict__ sn_a,
                               int Bb, int T, int H, int N, int selN)
{
    int tid = blockIdx.x * blockDim.x + threadIdx.x;
    if (tid >= Bb * H) return;
    const int b = tid / H, h = tid % H;
    const int scal0 = 2 * H * N;          // start of scal block in sel row
    float angle = 0.f;
    const float bias = dt_bias[h];
    for (int t = 0; t < T; ++t) {
        const size_t row = (size_t)(b * T + t) * selN + scal0;
        const float s0 = sel[row + h];
        const float s1 = sel[row + H + h];
        const float s2 = sel[row + 2 * H + h];
        const float xx = s0 + bias;
        const float dt = (xx > 20.f) ? xx : log1pf(expf(xx));
        const float lam = 1.f / (1.f + expf(-s1));
        angle += dt * s2;
        const size_t o = (size_t)(b * T + t) * H + h;
        dt_a[o] = dt; lam_a[o] = lam;
        cs_a[o] = cosf(angle); sn_a[o] = sinf(angle);
    }
}

// ---------------------------------------------------------------------------
// RoPE in-place on the B_proj / C_proj halves of sel. One angle per (b,t,h).
// ---------------------------------------------------------------------------
__global__ void rope_inplace(float* __restrict__ sel,
                             const float* __restrict__ cs_a,
                             const float* __restrict__ sn_a,
                             int MT, int H, int N, int selN)
{
    const int idx = blockIdx.x * blockDim.x + threadIdx.x;
    const int halfN = N / 2;
    const int total = MT * H * halfN;
    if (idx >= total) return;
    const int i = idx % halfN;
    const int h = (idx / halfN) % H;
    const int m = idx / (H * halfN);
    const float c = cs_a[(size_t)m * H + h];
    const float s = sn_a[(size_t)m * H + h];

    size_t base = (size_t)m * selN + (size_t)h * N + 2 * i;  // B_proj
    float z0 = sel[base], z1 = sel[base + 1];
    sel[base]     = z0 * c - z1 * s;
    sel[base + 1] = z0 * s + z1 * c;

    base += (size_t)H * N;                                   // C_proj
    z0 = sel[base]; z1 = sel[base + 1];
    sel[base]     = z0 * c - z1 * s;
    sel[base + 1] = z0 * s + z1 * c;
}

// ---------------------------------------------------------------------------
// Sequential SSM scan. Grid = B*H blocks; 512 threads keep the full 64x128
// fp32 state in registers (16 floats/thread). B/C/u staged via LDS per step;
// y_t[p] reduced over 8 lanes with shfl.
// ---------------------------------------------------------------------------
__global__ __launch_bounds__(512)
void ssm_scan(const float* __restrict__ u, const float* __restrict__ sel,
              const float* __restrict__ dt_a, const float* __restrict__ lam_a,
              const float* __restrict__ A_log, float* __restrict__ y,
              int Bb, int T, int H, int P, int N, int selN)
{
    __shared__ float sB[128], sC[128], sU[64];
    __shared__ float sDt, sLam;

    const int blk = blockIdx.x;        // (b,h)
    const int b = blk / H, h = blk % H;
    const int tid = threadIdx.x;
    const int p  = tid >> 3;           // 0..63
    const int j  = tid & 7;            // 8 lanes share one p
    const int n0 = j * 16;
    const int d_inner = H * P;

    const float A_neg = -expf(A_log[h]);

    float hreg[16];
    float bprev[16];
    #pragma unroll
    for (int i = 0; i < 16; ++i) { hreg[i] = 0.f; bprev[i] = 0.f; }
    float uprev = 0.f;

    for (int t = 0; t < T; ++t) {
        const size_t mrow = (size_t)(b * T + t);
        if (tid < N) {
            sB[tid] = sel[mrow * selN + (size_t)h * N + tid];
        } else if (tid < 2 * N) {
            sC[tid - N] = sel[mrow * selN + (size_t)H * N + (size_t)h * N + (tid - N)];
        } else if (tid < 2 * N + P) {
            sU[tid - 2 * N] = u[mrow * d_inner + h * P + (tid - 2 * N)];
        } else if (tid == 2 * N + P) {
            sDt  = dt_a[mrow * H + h];
            sLam = lam_a[mrow * H + h];
        }
        __syncthreads();

        const float dt    = sDt;
        const float lam   = sLam;
        const float alpha = __expf(dt * A_neg);
        const float beta  = (1.f - lam) * dt * alpha;
        const float gamma = lam * dt;
        const float ut    = sU[p];

        float partial = 0.f;
        #pragma unroll
        for (int i = 0; i < 16; ++i) {
            const float bn = sB[n0 + i];
            const float hn = alpha * hreg[i] + beta * (uprev * bprev[i]) + gamma * (ut * bn);
            hreg[i] = hn;
            partial += hn * sC[n0 + i];
            bprev[i] = bn;
        }
        uprev = ut;

        partial += __shfl_down(partial, 4, 8);
        partial += __shfl_down(partial, 2, 8);
        partial += __shfl_down(partial, 1, 8);
        if (j == 0) y[mrow * d_inner + h * P + p] = partial;
        __syncthreads();
    }
}

// ---------------------------------------------------------------------------
// Row-wise RMSNorm over d_inner.
// ---------------------------------------------------------------------------
__global__ __launch_bounds__(256)
void rmsnorm_rows(const float* __restrict__ y, const float* __restrict__ w,
                  float* __restrict__ out, int M, int D)
{
    __shared__ float red[8];
    const int m = blockIdx.x;
    const int tid = threadIdx.x;
    float ss = 0.f;
    for (int i = tid; i < D; i += blockDim.x) {
        const float v = y[(size_t)m * D + i];
        ss += v * v;
    }
    for (int off = 16; off > 0; off >>= 1) ss += __shfl_down(ss, off, 32);
    if ((tid & 31) == 0) red[tid >> 5] = ss;
    __syncthreads();
    if (tid == 0) {
        float tot = 0.f;
        for (int i = 0; i < (int)(blockDim.x >> 5); ++i) tot += red[i];
        red[0] = rsqrtf(tot / (float)D + 1e-6f);
    }
    __syncthreads();
    const float scale = red[0];
    for (int i = tid; i < D; i += blockDim.x)
        out[(size_t)m * D + i] = y[(size_t)m * D + i] * scale * w[i];
}

// ---------------------------------------------------------------------------
extern "C" void kernel_launch(void* const* d_in, const int* in_sizes, int n_in,
                              void* d_out, int out_size, void* d_ws, size_t ws_size,
                              hipStream_t stream) {
    (void)in_sizes; (void)n_in; (void)out_size; (void)ws_size;

    const float* x       = (const float*)d_in[0];
    const float* Win     = (const float*)d_in[1];
    const float* Wsel    = (const float*)d_in[2];
    const float* Wout    = (const float*)d_in[3];
    const float* A_log   = (const float*)d_in[4];
    const float* dt_bias = (const float*)d_in[5];
    const float* rms_w   = (const float*)d_in[6];
    float* out = (float*)d_out;

    constexpr int Bb = 2, T = 1024, D = 1024, H = 8, P = 64, N = 128;
    constexpr int M       = Bb * T;            // 2048
    constexpr int d_inner = H * P;             // 512
    constexpr int selN    = 2 * H * N + 3 * H; // 2072

    float* ws = (float*)d_ws;
    size_t off = 0;
    float* u_buf = ws + off; off += (size_t)M * d_inner;
    float* selb  = ws + off; off += (size_t)M * selN;
    float* dt_a  = ws + off; off += (size_t)M * H;
    float* lam_a = ws + off; off += (size_t)M * H;
    float* cs_a  = ws + off; off += (size_t)M * H;
    float* sn_a  = ws + off; off += (size_t)M * H;
    float* ybuf  = ws + off; off += (size_t)M * d_inner;
    float* ynorm = ws + off; off += (size_t)M * d_inner;

    // 1) u = x @ in_proj_w^T   (2048x1024 * 1024x512)
    gemm_xwt_f32<<<dim3(M / 64, d_inner / 64), dim3(256), 0, stream>>>(
        x, Win, u_buf, M, D, d_inner);
    // 2) sel = x @ sel_proj_w^T (2048x1024 * 1024x2072)
    gemm_xwt_f32<<<dim3(M / 64, (selN + 63) / 64), dim3(256), 0, stream>>>(
        x, Wsel, selb, M, D, selN);
    // 3) dt/lam + cumulative angle
    scalars_cumsum<<<1, Bb * H, 0, stream>>>(selb, dt_bias, dt_a, lam_a,
                                             cs_a, sn_a, Bb, T, H, N, selN);
    // 4) RoPE on B_proj/C_proj (in place in sel buffer)
    {
        const int total = M * H * (N / 2);
        rope_inplace<<<(total + 255) / 256, 256, 0, stream>>>(
            selb, cs_a, sn_a, M, H, N, selN);
    }
    // 5) sequential scan -> y
    ssm_scan<<<Bb * H, 512, 0, stream>>>(u_buf, selb, dt_a, lam_a, A_log, ybuf,
                                         Bb, T, H, P, N, selN);
    // 6) RMSNorm
    rmsnorm_rows<<<M, 256, 0, stream>>>(ybuf, rms_w, ynorm, M, d_inner);
    // 7) out = ynorm @ out_proj_w^T (2048x512 * 512x1024)
    gemm_xwt_f32<<<dim3(M / 64, D / 64), dim3(256), 0, stream>>>(
        ynorm, Wout, out, M, d_inner, D);
}